// MeanEmbedding_55525337202981
// MI455X (gfx1250) — compile-verified
//
#include <hip/hip_runtime.h>
#include <stdint.h>

// Problem constants (match reference).
constexpr int kVocab = 50257;
constexpr int kEmbed = 1024;   // D
constexpr int kL     = 2048;   // sequence length
constexpr int kB     = 64;     // batch
constexpr int kChunks = 8;                         // L-chunks per batch row
constexpr int kTokPerBlock = kL / kChunks;         // 256 tokens per block
constexpr int kWaves = 8;                          // 256 threads / wave32
constexpr int kTokPerWave = kTokPerBlock / kWaves; // 32 tokens per wave
constexpr int kDepth = 4;                          // async ring depth per wave

// ---------------------------------------------------------------------------
// Gather + masked partial-sum kernel.
// One block = (batch b, chunk c). Each of the 8 waves independently streams
// its 32 token rows through a private 4-deep ring of 4KB LDS stage buffers
// using CDNA5 async global->LDS loads (up to 16KB outstanding per wave in steady
// state), accumulating the full 1024-feature partial in 32 VGPRs. The stage
// LDS is then reused for a fixed-order (deterministic) cross-wave reduction.
// ---------------------------------------------------------------------------
template <bool USE_WS>
__global__ __launch_bounds__(256)
void gather_sum_kernel(const int* __restrict__ xs,
                       const int* __restrict__ xs_len,
                       const float* __restrict__ W,
                       float* __restrict__ dst /* ws (USE_WS) or out */) {
  __shared__ __align__(16) float stage[kWaves][kDepth][kEmbed];  // 128 KB

  const int b     = (int)blockIdx.x;
  const int chunk = (int)blockIdx.y;
  const int tid   = (int)threadIdx.x;
  const int wave  = tid >> 5;
  const int lane  = tid & 31;

  const int len  = xs_len[b];
  const int tok0 = chunk * kTokPerBlock + wave * kTokPerWave;
  int n = len - tok0;
  n = n < 0 ? 0 : (n > kTokPerWave ? kTokPerWave : n);

  float4 acc[8];
#pragma unroll
  for (int k = 0; k < 8; ++k) acc[k] = make_float4(0.f, 0.f, 0.f, 0.f);

  // LDS byte address of this wave's ring (low 32 bits of the flat shared
  // address are the LDS byte offset on gfx1250), +16B per lane.
  const uint32_t lds_base =
      (uint32_t)(uintptr_t)(&stage[wave][0][0]) + (uint32_t)(lane * 16);

  const int* tokp = xs + b * kL + tok0;

  // Stream one 4KB embedding row into an LDS stage buffer.
  // Each lane issues 8 async b128 loads (32 lanes * 16B * 8 = 4096B).
  // The instruction offset advances BOTH the global and the LDS address
  // (CDNA5 ISA pseudocode), so one base pair covers the row.
  auto issue_row = [&](int tok, uint32_t lds_addr) {
    uint64_t ga = (uint64_t)(uintptr_t)W +
                  (uint64_t)(uint32_t)tok * (uint64_t)(kEmbed * 4) +
                  (uint64_t)(uint32_t)(lane * 16);
    // Make sure our own earlier ds_loads of this buffer have completed
    // before the async engine may overwrite it.
    asm volatile("s_wait_dscnt 0" ::: "memory");
    asm volatile(
        "global_load_async_to_lds_b128 %0, %1, off\n\t"
        "global_load_async_to_lds_b128 %0, %1, off offset:512\n\t"
        "global_load_async_to_lds_b128 %0, %1, off offset:1024\n\t"
        "global_load_async_to_lds_b128 %0, %1, off offset:1536\n\t"
        "global_load_async_to_lds_b128 %0, %1, off offset:2048\n\t"
        "global_load_async_to_lds_b128 %0, %1, off offset:2560\n\t"
        "global_load_async_to_lds_b128 %0, %1, off offset:3072\n\t"
        "global_load_async_to_lds_b128 %0, %1, off offset:3584"
        :
        : "v"(lds_addr), "v"(ga)
        : "memory");
  };

  // Prologue: fill up to kDepth-1 = 3 buffers ahead.
  const int pre = n < (kDepth - 1) ? n : (kDepth - 1);
  for (int j = 0; j < pre; ++j)
    issue_row(tokp[j], lds_base + ((uint32_t)j << 12));

  // Prefetch next token id (uniform SMEM load) one iteration early so it is
  // covered by the accumulate phase instead of stalling the issue path.
  int tnext = ((kDepth - 1) < n) ? tokp[kDepth - 1] : 0;

  for (int i = 0; i < n; ++i) {
    const int j = i + (kDepth - 1);
    const int tcur = tnext;
    tnext = (j + 1 < n) ? tokp[j + 1] : 0;
    if (j < n) issue_row(tcur, lds_base + ((uint32_t)(j & (kDepth - 1)) << 12));

    // Async loads complete in order per wave; to guarantee row i has landed,
    // allow 8 outstanding instructions per not-yet-needed younger row.
    int ahead = n - 1 - i;
    ahead = ahead > (kDepth - 1) ? (kDepth - 1) : ahead;
    switch (ahead) {
      case 3:  asm volatile("s_wait_asynccnt 24" ::: "memory"); break;
      case 2:  asm volatile("s_wait_asynccnt 16" ::: "memory"); break;
      case 1:  asm volatile("s_wait_asynccnt 8"  ::: "memory"); break;
      default: asm volatile("s_wait_asynccnt 0"  ::: "memory"); break;
    }

    const float4* buf = (const float4*)(&stage[wave][i & (kDepth - 1)][0]);
#pragma unroll
    for (int k = 0; k < 8; ++k) {
      float4 v = buf[k * 32 + lane];   // feature k*128 + lane*4 .. +3
      acc[k].x += v.x; acc[k].y += v.y; acc[k].z += v.z; acc[k].w += v.w;
    }
  }

  // Publish this wave's 1024-feature partial (zeros if n==0) into its own
  // buffer 0 (safe: only this wave ever touched it, and asynccnt==0 here).
#pragma unroll
  for (int k = 0; k < 8; ++k)
    ((float4*)&stage[wave][0][0])[k * 32 + lane] = acc[k];
  __syncthreads();

  // Fixed-order (deterministic) cross-wave reduction: thread t owns
  // features 4t..4t+3.
  float4 s = make_float4(0.f, 0.f, 0.f, 0.f);
#pragma unroll
  for (int w = 0; w < kWaves; ++w) {
    float4 v = ((const float4*)&stage[w][0][0])[tid];
    s.x += v.x; s.y += v.y; s.z += v.z; s.w += v.w;
  }

  if (USE_WS) {
    float4* o = (float4*)(dst + (size_t)(b * kChunks + chunk) * kEmbed) + tid;
    *o = s;
  } else {
    const float inv_len = 1.0f / (float)len;
    float* o = dst + (size_t)b * kEmbed + tid * 4;
    atomicAdd(o + 0, s.x * inv_len);
    atomicAdd(o + 1, s.y * inv_len);
    atomicAdd(o + 2, s.z * inv_len);
    atomicAdd(o + 3, s.w * inv_len);
  }
}

// Sum the 8 chunk partials per batch row and apply 1/len. Deterministic.
__global__ __launch_bounds__(256)
void finalize_kernel(const float* __restrict__ ws,
                     const int* __restrict__ xs_len,
                     float* __restrict__ out) {
  const int idx = (int)(blockIdx.x * blockDim.x + threadIdx.x); // float4 index
  const int total = kB * kEmbed / 4;
  if (idx >= total) return;
  const int b = idx >> 8;           // kEmbed/4 == 256 float4 per row
  const int r = idx & 255;

  float4 s = make_float4(0.f, 0.f, 0.f, 0.f);
#pragma unroll
  for (int c = 0; c < kChunks; ++c) {
    float4 v = ((const float4*)ws)[(size_t)(b * kChunks + c) * 256 + r];
    s.x += v.x; s.y += v.y; s.z += v.z; s.w += v.w;
  }
  const float inv = 1.0f / (float)xs_len[b];
  float4 o = make_float4(s.x * inv, s.y * inv, s.z * inv, s.w * inv);
  ((float4*)out)[idx] = o;
}

__global__ void zero_kernel(float* __restrict__ p, int n) {
  const int i = (int)(blockIdx.x * blockDim.x + threadIdx.x);
  if (i < n) p[i] = 0.f;
}

extern "C" void kernel_launch(void* const* d_in, const int* in_sizes, int n_in,
                              void* d_out, int out_size, void* d_ws, size_t ws_size,
                              hipStream_t stream) {
  (void)in_sizes; (void)n_in;
  const int*   xs     = (const int*)d_in[0];    // [B, L] token ids
  const int*   xs_len = (const int*)d_in[1];    // [B]
  const float* W      = (const float*)d_in[2];  // [VOCAB, EMBED] f32
  float*       out    = (float*)d_out;          // [B, EMBED] f32

  const size_t ws_needed = (size_t)kB * kChunks * kEmbed * sizeof(float); // 2 MB
  dim3 grid(kB, kChunks);

  if (ws_size >= ws_needed) {
    float* ws = (float*)d_ws;
    gather_sum_kernel<true><<<grid, 256, 0, stream>>>(xs, xs_len, W, ws);
    const int nf4 = kB * kEmbed / 4;
    finalize_kernel<<<(nf4 + 255) / 256, 256, 0, stream>>>(ws, xs_len, out);
  } else {
    zero_kernel<<<(out_size + 255) / 256, 256, 0, stream>>>(out, out_size);
    gather_sum_kernel<false><<<grid, 256, 0, stream>>>(xs, xs_len, W, out);
  }
}